// DF_56985626083519
// MI455X (gfx1250) — compile-verified
//
#include <hip/hip_runtime.h>
#include <math.h>

typedef __attribute__((ext_vector_type(2))) float v2f;
typedef __attribute__((ext_vector_type(8))) float v8f;

__device__ __forceinline__ float bilin(const float* __restrict__ ch, int Hs, int Ws,
                                       float sy, float sx) {
    float fy = floorf(sy), fx = floorf(sx);
    int y0 = (int)fy, x0 = (int)fx;
    float wy = sy - fy, wx = sx - fx;
    int y0c = min(max(y0, 0), Hs - 1);
    int y1c = min(max(y0 + 1, 0), Hs - 1);
    int x0c = min(max(x0, 0), Ws - 1);
    int x1c = min(max(x0 + 1, 0), Ws - 1);
    float v00 = ch[y0c * Ws + x0c], v01 = ch[y0c * Ws + x1c];
    float v10 = ch[y1c * Ws + x0c], v11 = ch[y1c * Ws + x1c];
    return (1.f - wy) * ((1.f - wx) * v00 + wx * v01) +
           wy * ((1.f - wx) * v10 + wx * v11);
}

__device__ __forceinline__ float gelu_exact(float v) {
    return 0.5f * v * (1.f + erff(v * 0.70710678118654752f));
}

// ---------------------------------------------------------------------------
// Bilinear resize (half-pixel centers, edge clamp)
// ---------------------------------------------------------------------------
__global__ void resize_bilinear(const float* __restrict__ src, float* __restrict__ dst,
                                int C, int Hs, int Ws, int Hd, int Wd) {
    int idx = blockIdx.x * blockDim.x + threadIdx.x;
    int total = C * Hd * Wd;
    if (idx >= total) return;
    int c = idx / (Hd * Wd);
    int p = idx % (Hd * Wd);
    int y = p / Wd, x = p % Wd;
    float sy = (y + 0.5f) * (float)Hs / (float)Hd - 0.5f;
    float sx = (x + 0.5f) * (float)Ws / (float)Wd - 0.5f;
    dst[idx] = bilin(src + (size_t)c * Hs * Ws, Hs, Ws, sy, sx);
}

// ---------------------------------------------------------------------------
// Offset conv: 3x3, 64 -> 18 channels, pad=1 dil=1, via WMMA f32 16x16x4.
// H is a compile-time constant so channel strides fold into immediate load
// offsets; OOB handled by clamped index + 0/1 mask multiply (no EXEC branching).
// ---------------------------------------------------------------------------
template <int H>
__global__ void __launch_bounds__(128)
conv3x3_off_wmma(const float* __restrict__ src, const float* __restrict__ w,
                 const float* __restrict__ bias, float* __restrict__ dst) {
    constexpr int W = H, HW = H * W;
    __shared__ float sW[576 * 32];                 // [K][co_pad32], K = tap*64 + ci
    for (int t = threadIdx.x; t < 576 * 32; t += 128) {
        int K = t >> 5, co = t & 31;
        int k = K >> 6, c = K & 63;
        sW[t] = (co < 18) ? w[co * 576 + c * 9 + k] : 0.f;
    }
    __syncthreads();

    const int lane = threadIdx.x & 31, wv = threadIdx.x >> 5;
    const int m = lane & 15, half = lane >> 4;
    const int coBase = blockIdx.y * 16;
    const int p0 = (blockIdx.x * 4 + wv) * 16;
    const int p = p0 + m, py = p / W, px = p % W;

    v8f acc = {};
    for (int k = 0; k < 9; ++k) {
        int yy = py + (k / 3) - 1;
        int xx = px + (k % 3) - 1;
        float msk = ((yy >= 0) & (yy < H) & (xx >= 0) & (xx < W)) ? 1.f : 0.f;
        int bi = min(max(yy * W + xx, 0), HW - 1);
        const float* sp = src + bi + half * (2 * HW);     // per-lane base; jj offsets are immediates
        const float* sb = sW + k * 64 * 32 + coBase + m + half * (2 * 32);
#pragma unroll
        for (int jj = 0; jj < 16; ++jj) {
            v2f a, b;
            a[0] = msk * sp[(jj * 4 + 0) * HW];
            a[1] = msk * sp[(jj * 4 + 1) * HW];
            b[0] = sb[(jj * 4 + 0) * 32];
            b[1] = sb[(jj * 4 + 1) * 32];
            acc = __builtin_amdgcn_wmma_f32_16x16x4_f32(false, a, false, b,
                                                        (short)0, acc, false, false);
        }
    }
    int co = coBase + m;
    if (co < 18) {
        float bv = bias[co];
#pragma unroll
        for (int r = 0; r < 8; ++r) {
            int pr = p0 + r + 8 * half;
            dst[(size_t)co * HW + pr] = acc[r] + bv;
        }
    }
}

// ---------------------------------------------------------------------------
// Deformable conv: 9 taps, 64 -> 16 channels, bilinear sampling, fused leaky.
// Per-wave tap tables (clamped indices + mask-folded weights) in LDS, reused
// over all 64 channels; weights staged reordered in LDS. H/PAD/DIL static.
// ---------------------------------------------------------------------------
template <int H, int PAD, int DIL>
__global__ void __launch_bounds__(128)
deform_conv_wmma(const float* __restrict__ src, const float* __restrict__ off,
                 const float* __restrict__ w, const float* __restrict__ bias,
                 float* __restrict__ dst) {
    constexpr int W = H, HW = H * W;
    __shared__ float sW[576 * 16];                 // [K][co], K = tap*64 + ci
    __shared__ int   sTI[4][144 * 4];
    __shared__ float sTW[4][144 * 4];

    for (int t = threadIdx.x; t < 576 * 16; t += 128) {
        int K = t >> 4, n = t & 15;
        int k = K >> 6, c = K & 63;
        sW[t] = w[n * 576 + c * 9 + k];
    }

    const int lane = threadIdx.x & 31, wv = threadIdx.x >> 5;
    const int m = lane & 15, half = lane >> 4;
    const int p0 = (blockIdx.x * 4 + wv) * 16;

    // Tap precompute: 16 pixels x 9 taps = 144 jobs per wave
    for (int t = lane; t < 144; t += 32) {
        int mm = t / 9, k = t % 9;
        int p = p0 + mm, yi = p / W, xi = p % W;
        float dy = off[(size_t)(2 * k) * HW + p];
        float dx = off[(size_t)(2 * k + 1) * HW + p];
        float py = (float)(yi - PAD + (k / 3) * DIL) + dy;
        float px = (float)(xi - PAD + (k % 3) * DIL) + dx;
        float fy = floorf(py), fx = floorf(px);
        int y0 = (int)fy, x0 = (int)fx;
        float wy = py - fy, wx = px - fx;
        float vy0 = (y0 >= 0 && y0 <= H - 1) ? 1.f : 0.f;
        float vy1 = (y0 + 1 >= 0 && y0 + 1 <= H - 1) ? 1.f : 0.f;
        float vx0 = (x0 >= 0 && x0 <= W - 1) ? 1.f : 0.f;
        float vx1 = (x0 + 1 >= 0 && x0 + 1 <= W - 1) ? 1.f : 0.f;
        int y0c = min(max(y0, 0), H - 1), y1c = min(max(y0 + 1, 0), H - 1);
        int x0c = min(max(x0, 0), W - 1), x1c = min(max(x0 + 1, 0), W - 1);
        int tb = t * 4;
        sTI[wv][tb + 0] = y0c * W + x0c;
        sTI[wv][tb + 1] = y0c * W + x1c;
        sTI[wv][tb + 2] = y1c * W + x0c;
        sTI[wv][tb + 3] = y1c * W + x1c;
        sTW[wv][tb + 0] = (1.f - wy) * (1.f - wx) * vy0 * vx0;
        sTW[wv][tb + 1] = (1.f - wy) * wx * vy0 * vx1;
        sTW[wv][tb + 2] = wy * (1.f - wx) * vy1 * vx0;
        sTW[wv][tb + 3] = wy * wx * vy1 * vx1;
    }
    __syncthreads();

    v8f acc = {};
    for (int k = 0; k < 9; ++k) {
        int tb = (m * 9 + k) * 4;
        int i00 = sTI[wv][tb + 0], i01 = sTI[wv][tb + 1];
        int i10 = sTI[wv][tb + 2], i11 = sTI[wv][tb + 3];
        float w00 = sTW[wv][tb + 0], w01 = sTW[wv][tb + 1];
        float w10 = sTW[wv][tb + 2], w11 = sTW[wv][tb + 3];
        const int hofs = half * (2 * HW);
        const float* b00 = src + i00 + hofs;
        const float* b01 = src + i01 + hofs;
        const float* b10 = src + i10 + hofs;
        const float* b11 = src + i11 + hofs;
        const float* sb = sW + k * 64 * 16 + m + half * (2 * 16);
#pragma unroll
        for (int jj = 0; jj < 16; ++jj) {
            constexpr int C0 = 0;
            int o0 = (jj * 4 + 0) * HW, o1 = (jj * 4 + 1) * HW;
            v2f a, b;
            a[0] = w00 * b00[o0] + w01 * b01[o0] + w10 * b10[o0] + w11 * b11[o0];
            a[1] = w00 * b00[o1] + w01 * b01[o1] + w10 * b10[o1] + w11 * b11[o1];
            b[0] = sb[(jj * 4 + 0) * 16];
            b[1] = sb[(jj * 4 + 1) * 16];
            acc = __builtin_amdgcn_wmma_f32_16x16x4_f32(false, a, false, b,
                                                        (short)0, acc, false, false);
            (void)C0;
        }
    }
    float bv = bias[m];
#pragma unroll
    for (int r = 0; r < 8; ++r) {
        int pr = p0 + r + 8 * half;
        float v = acc[r] + bv;
        dst[(size_t)m * HW + pr] = (v > 0.f) ? v : 0.01f * v;
    }
}

// ---------------------------------------------------------------------------
// Final: resize a1/a2/a3 back to 64x64 inline + gelu combine -> 4 outputs
// ---------------------------------------------------------------------------
__global__ void final_combine(const float* __restrict__ a,
                              const float* __restrict__ a1,
                              const float* __restrict__ a2,
                              const float* __restrict__ a3,
                              float* __restrict__ out) {
    int idx = blockIdx.x * blockDim.x + threadIdx.x;
    if (idx >= 16 * 64 * 64) return;
    int c = idx >> 12, p = idx & 4095;
    int y = p >> 6, x = p & 63;
    float x1v = bilin(a1 + (size_t)c * 32 * 32, 32, 32,
                      (y + 0.5f) * 0.5f - 0.5f, (x + 0.5f) * 0.5f - 0.5f);
    float x2v = bilin(a2 + (size_t)c * 128 * 128, 128, 128,
                      (y + 0.5f) * 2.f - 0.5f, (x + 0.5f) * 2.f - 0.5f);
    float x3v = bilin(a3 + (size_t)c * 256 * 256, 256, 256,
                      (y + 0.5f) * 4.f - 0.5f, (x + 0.5f) * 4.f - 0.5f);
    float av = a[idx];
    float l = gelu_exact(x1v);
    float mo = gelu_exact(av - l);
    float h = gelu_exact(x2v - av);
    float so = gelu_exact(x3v - x2v);
    out[idx] = l;
    out[65536 + idx] = mo;
    out[131072 + idx] = h;
    out[196608 + idx] = so;
}

// ---------------------------------------------------------------------------
extern "C" void kernel_launch(void* const* d_in, const int* in_sizes, int n_in,
                              void* d_out, int out_size, void* d_ws, size_t ws_size,
                              hipStream_t stream) {
    const float* x = (const float*)d_in[0];
    const float* w_off[4]; const float* b_off[4];
    const float* w_c[4];   const float* b_c[4];
    for (int i = 0; i < 4; ++i) {
        w_off[i] = (const float*)d_in[1 + 4 * i];
        b_off[i] = (const float*)d_in[2 + 4 * i];
        w_c[i]   = (const float*)d_in[3 + 4 * i];
        b_c[i]   = (const float*)d_in[4 + 4 * i];
    }
    float* ws = (float*)d_ws;
    float* x1   = ws; ws += 64 * 32 * 32;
    float* x2   = ws; ws += 64 * 128 * 128;
    float* x3   = ws; ws += (size_t)64 * 256 * 256;
    float* off0 = ws; ws += 18 * 64 * 64;
    float* off1 = ws; ws += 18 * 32 * 32;
    float* off2 = ws; ws += 18 * 128 * 128;
    float* off3 = ws; ws += (size_t)18 * 256 * 256;
    float* a    = ws; ws += 16 * 64 * 64;
    float* a1   = ws; ws += 16 * 32 * 32;
    float* a2   = ws; ws += 16 * 128 * 128;
    float* a3   = ws; ws += (size_t)16 * 256 * 256;
    float* out = (float*)d_out;

    // resizes of the input pyramid
    {
        int t1 = 64 * 32 * 32;
        resize_bilinear<<<(t1 + 255) / 256, 256, 0, stream>>>(x, x1, 64, 64, 64, 32, 32);
        int t2 = 64 * 128 * 128;
        resize_bilinear<<<(t2 + 255) / 256, 256, 0, stream>>>(x, x2, 64, 64, 64, 128, 128);
        int t3 = 64 * 256 * 256;
        resize_bilinear<<<(t3 + 255) / 256, 256, 0, stream>>>(x2, x3, 64, 128, 128, 256, 256);
    }

    // offset convs (3x3, 18 channels) — WMMA
    conv3x3_off_wmma<32><<<dim3(32 * 32 / 64, 2), 128, 0, stream>>>(x1, w_off[0], b_off[0], off1);
    conv3x3_off_wmma<64><<<dim3(64 * 64 / 64, 2), 128, 0, stream>>>(x, w_off[1], b_off[1], off0);
    conv3x3_off_wmma<128><<<dim3(128 * 128 / 64, 2), 128, 0, stream>>>(x2, w_off[2], b_off[2], off2);
    conv3x3_off_wmma<256><<<dim3(256 * 256 / 64, 2), 128, 0, stream>>>(x3, w_off[3], b_off[3], off3);

    // deformable convs (fused leaky-relu) — WMMA
    deform_conv_wmma<32, 4, 4><<<dim3(32 * 32 / 64), 128, 0, stream>>>(x1, off1, w_c[3], b_c[3], a1);
    deform_conv_wmma<64, 3, 3><<<dim3(64 * 64 / 64), 128, 0, stream>>>(x, off0, w_c[2], b_c[2], a);
    deform_conv_wmma<128, 2, 2><<<dim3(128 * 128 / 64), 128, 0, stream>>>(x2, off2, w_c[1], b_c[1], a2);
    deform_conv_wmma<256, 1, 1><<<dim3(256 * 256 / 64), 128, 0, stream>>>(x3, off3, w_c[0], b_c[0], a3);

    // fused resize-back + gelu combine
    final_combine<<<(16 * 64 * 64 + 255) / 256, 256, 0, stream>>>(a, a1, a2, a3, out);
}